// LSTMLayer_12584254177428
// MI455X (gfx1250) — compile-verified
//
#include <hip/hip_runtime.h>

// ---------------------------------------------------------------------------
// DGM gated layer on MI455X (gfx1250), fp32 WMMA path (V_WMMA_F32_16X16X4_F32)
//   Z = tanh(X@Uz + S@Wz + bz)
//   G = tanh(X@Ug + S@Wg + bg)
//   R = tanh(X@Ur + S@Wr + br)
//   H = tanh(X@Uh + (S*R)@Wh + bh)
//   S_new = (1-G)*H + Z*S
// Persistent wave32s grid-stride over 16-row batch tiles; OUT_DIM=64 ->
// 4 N-tiles of 16, processed in pairs with all 2x17 B-fragments preloaded
// (deep load clauses, extended VGPR file) and two interleaved WMMA chains.
// Next tile's S/X rows are prefetched (global_prefetch_b8) so HBM latency
// overlaps the current tile's 272 WMMAs. tanh = v_tanh_f32 TRANS op.
// ---------------------------------------------------------------------------

typedef __attribute__((ext_vector_type(2))) float v2f;
typedef __attribute__((ext_vector_type(8))) float v8f;

#define IN_DIM            3
#define OUT_DIM           64
#define ROWS_PER_WAVE     16
#define WAVES_PER_BLOCK   4      // 1 wave/SIMD: placement always legal
#define PERSISTENT_BLOCKS 1024   // ~16 tiles per wave at batch = 1M

__device__ __forceinline__ float fast_tanh(float x) {
#if __has_builtin(__builtin_amdgcn_tanhf)
    return __builtin_amdgcn_tanhf(x);        // v_tanh_f32 (gfx1250 TRANS op)
#elif __has_builtin(__builtin_amdgcn_tanh_f32)
    return __builtin_amdgcn_tanh_f32(x);
#else
    return tanhf(x);
#endif
}

__device__ __forceinline__ v8f wmma_k4(v2f a, v2f b, v8f c) {
    // 8 args: (neg_a, A, neg_b, B, c_mod, C, reuse_a, reuse_b)
    return __builtin_amdgcn_wmma_f32_16x16x4_f32(false, a, false, b,
                                                 (short)0, c, false, false);
}

// out[n] = bias + X@U + A@W  for one gate, over the 4 N-tiles (paired).
// aK: 16 A-fragments (16x4 each) of the 16x64 left operand.
__device__ __forceinline__ void gate_matmul(const v2f aK[16], v2f aX,
                                            const float* __restrict__ U,
                                            const float* __restrict__ W,
                                            const float* __restrict__ bias,
                                            int hi, int lr, v8f out[4]) {
#pragma unroll
    for (int np = 0; np < 2; ++np) {
        const int c0 = (2 * np) * 16 + lr;        // cols of N-tile pair
        const int c1 = (2 * np + 1) * 16 + lr;

        // ---- preload ALL B fragments for both tiles: 68 distinct VGPRs ----
        // U B-frag 4x16 (K=3 zero-padded): VGPR0 = rows {0,2}, VGPR1 = {1,pad}
        v2f bu0, bu1;
        bu0.x = U[(hi ? 2 : 0) * OUT_DIM + c0];
        bu1.x = U[(hi ? 2 : 0) * OUT_DIM + c1];
        const float u10 = U[1 * OUT_DIM + c0];    // always-valid row 1
        const float u11 = U[1 * OUT_DIM + c1];
        bu0.y = hi ? 0.0f : u10;                  // branch-free zero pad
        bu1.y = hi ? 0.0f : u11;

        v2f bw0[16], bw1[16];
#pragma unroll
        for (int k = 0; k < 16; ++k) {
            const int rb = 4 * k + 2 * hi;        // base row of this lane-half
            bw0[k].x = W[rb * OUT_DIM + c0];
            bw0[k].y = W[(rb + 1) * OUT_DIM + c0];
            bw1[k].x = W[rb * OUT_DIM + c1];
            bw1[k].y = W[(rb + 1) * OUT_DIM + c1];
        }

        // ---- two independent accumulator chains, bias-initialized ----
        const float bv0 = bias[c0];
        const float bv1 = bias[c1];
        v8f a0, a1;
#pragma unroll
        for (int r = 0; r < 8; ++r) { a0[r] = bv0; a1[r] = bv1; }

        a0 = wmma_k4(aX, bu0, a0);                // X @ U (K padded 3->4)
        a1 = wmma_k4(aX, bu1, a1);
#pragma unroll
        for (int k = 0; k < 16; ++k) {            // A(16x64) @ W block
            a0 = wmma_k4(aK[k], bw0[k], a0);
            a1 = wmma_k4(aK[k], bw1[k], a1);
        }
        out[2 * np]     = a0;
        out[2 * np + 1] = a1;
    }
}

__global__ __launch_bounds__(WAVES_PER_BLOCK * 32, 1)
void dgm_layer_kernel(const float* __restrict__ S,  const float* __restrict__ X,
                      const float* __restrict__ Uz, const float* __restrict__ Ug,
                      const float* __restrict__ Ur, const float* __restrict__ Uh,
                      const float* __restrict__ Wz, const float* __restrict__ Wg,
                      const float* __restrict__ Wr, const float* __restrict__ Wh,
                      const float* __restrict__ bz, const float* __restrict__ bg,
                      const float* __restrict__ br, const float* __restrict__ bh,
                      float* __restrict__ Sout, int batch) {
    __shared__ float ldsT[WAVES_PER_BLOCK * ROWS_PER_WAVE * OUT_DIM];  // 16 KB

    const int lane = threadIdx.x & 31;
    const int wib  = threadIdx.x >> 5;
    const int hi   = lane >> 4;                  // lane half: 0 | 1
    const int lr   = lane & 15;                  // A-row / B,C,D-col index
    float* tbuf = &ldsT[wib * ROWS_PER_WAVE * OUT_DIM];

    const int ntiles = (batch + ROWS_PER_WAVE - 1) / ROWS_PER_WAVE;
    const int step   = gridDim.x * WAVES_PER_BLOCK;     // wave grid stride

    for (int tile = blockIdx.x * WAVES_PER_BLOCK + wib; tile < ntiles;
         tile += step) {
        const int m0 = tile * ROWS_PER_WAVE;

        // ---- prefetch next tile's S (4 KB) and X rows into the caches ----
        const int pm0 = m0 + step * ROWS_PER_WAVE;      // wave-uniform
        if (pm0 < batch) {
            const char* ps = (const char*)(S + (size_t)pm0 * OUT_DIM);
            __builtin_prefetch(ps + lane * 128, 0, 3);  // 32 x 128B = 4 KB
            const char* px = (const char*)(X + (size_t)pm0 * IN_DIM);
            __builtin_prefetch(px + (lane & 1) * 128, 0, 3);   // 192 B
        }

        // ---- S tile, WMMA-A layout: lanes0-15 K={0,1}, lanes16-31 K={2,3} --
        v2f aS[16];
#pragma unroll
        for (int k = 0; k < 16; ++k)
            aS[k] = *(const v2f*)(S + (size_t)(m0 + lr) * OUT_DIM +
                                  4 * k + 2 * hi);

        // ---- S tile, C/D layout: VGPR r -> row r (lo half) / r+8 (hi half) -
        v8f sD[4];
#pragma unroll
        for (int n = 0; n < 4; ++n) {
            const int col = n * 16 + lr;
#pragma unroll
            for (int r = 0; r < 8; ++r)
                sD[n][r] = S[(size_t)(m0 + r + 8 * hi) * OUT_DIM + col];
        }

        // ---- X tile, A layout, K padded 3->4 with zeros (branch-free) ----
        v2f aX;
        {
            const float* xp = X + (size_t)(m0 + lr) * IN_DIM;
            const float x_even = xp[2 * hi];   // K=0 (lo half) | K=2 (hi half)
            const float x_odd  = xp[1];        // always-valid col 1
            aX.x = x_even;
            aX.y = hi ? 0.0f : x_odd;          // K=1 | K=3 zero pad
        }

        v8f tf[4], zf[4], gf[4], hf[4];

        // ---- R gate, then T = S * tanh(R) staged to LDS (row-major 16x64) --
        gate_matmul(aS, aX, Ur, Wr, br, hi, lr, tf);
#pragma unroll
        for (int n = 0; n < 4; ++n) {
            const int col = n * 16 + lr;
#pragma unroll
            for (int r = 0; r < 8; ++r)
                tbuf[(r + 8 * hi) * OUT_DIM + col] =
                    sD[n][r] * fast_tanh(tf[n][r]);
        }

        // ---- Z and G gates (reuse the resident aS fragments) ----
        gate_matmul(aS, aX, Uz, Wz, bz, hi, lr, zf);
        gate_matmul(aS, aX, Ug, Wg, bg, hi, lr, gf);
#pragma unroll
        for (int n = 0; n < 4; ++n)
#pragma unroll
            for (int r = 0; r < 8; ++r) {
                zf[n][r] = fast_tanh(zf[n][r]);
                gf[n][r] = fast_tanh(gf[n][r]);
            }

        // ---- read T back in A layout (wave-private LDS; drain DS first) ----
        asm volatile("s_wait_dscnt 0" ::: "memory");
        v2f aT[16];
#pragma unroll
        for (int k = 0; k < 16; ++k)
            aT[k] = *(const v2f*)&tbuf[lr * OUT_DIM + 4 * k + 2 * hi];

        // ---- H gate ----
        gate_matmul(aT, aX, Uh, Wh, bh, hi, lr, hf);

        // ---- S_new = (1-G)*H + Z*S, stored in C/D layout (coalesced, NT) --
#pragma unroll
        for (int n = 0; n < 4; ++n) {
            const int col = n * 16 + lr;
#pragma unroll
            for (int r = 0; r < 8; ++r) {
                const float h = fast_tanh(hf[n][r]);
                const float v = (1.0f - gf[n][r]) * h + zf[n][r] * sD[n][r];
                __builtin_nontemporal_store(
                    v, &Sout[(size_t)(m0 + r + 8 * hi) * OUT_DIM + col]);
            }
        }
    }
}

extern "C" void kernel_launch(void* const* d_in, const int* in_sizes, int n_in,
                              void* d_out, int out_size, void* d_ws, size_t ws_size,
                              hipStream_t stream) {
    (void)n_in; (void)d_ws; (void)ws_size; (void)out_size;
    const float* S  = (const float*)d_in[0];
    const float* X  = (const float*)d_in[1];
    const float* Uz = (const float*)d_in[2];
    const float* Ug = (const float*)d_in[3];
    const float* Ur = (const float*)d_in[4];
    const float* Uh = (const float*)d_in[5];
    const float* Wz = (const float*)d_in[6];
    const float* Wg = (const float*)d_in[7];
    const float* Wr = (const float*)d_in[8];
    const float* Wh = (const float*)d_in[9];
    const float* bz = (const float*)d_in[10];
    const float* bg = (const float*)d_in[11];
    const float* br = (const float*)d_in[12];
    const float* bh = (const float*)d_in[13];
    float* Sout = (float*)d_out;

    const int batch  = in_sizes[0] / OUT_DIM;                    // 1,000,000
    const int ntiles = (batch + ROWS_PER_WAVE - 1) / ROWS_PER_WAVE;
    const int full   = (ntiles + WAVES_PER_BLOCK - 1) / WAVES_PER_BLOCK;
    const int blocks = full < PERSISTENT_BLOCKS ? full : PERSISTENT_BLOCKS;

    dgm_layer_kernel<<<blocks, WAVES_PER_BLOCK * 32, 0, stream>>>(
        S, X, Uz, Ug, Ur, Uh, Wz, Wg, Wr, Wh, bz, bg, br, bh, Sout, batch);
}